// GCModule_33913061769301
// MI455X (gfx1250) — compile-verified
//
#include <hip/hip_runtime.h>

typedef __attribute__((ext_vector_type(2))) float v2f;
typedef __attribute__((ext_vector_type(8))) float v8f;

#define NNODES 50000
#define NEDGES 640000
#define INF    128
#define OUTF   128

// ---------------------------------------------------------------------------
// Kernel 1: zero the aggregation workspace (25.6 MB in d_ws)
// ---------------------------------------------------------------------------
__global__ void zero_ws(float4* __restrict__ agg4) {
  size_t i = (size_t)blockIdx.x * blockDim.x + threadIdx.x;
  agg4[i] = make_float4(0.f, 0.f, 0.f, 0.f);
}

// ---------------------------------------------------------------------------
// Kernel 2: edge gather + segment-sum via f32 atomics (resolve in 192MB L2).
// 32 lanes per edge, float4 gather (512B contiguous per wave per edge).
// ---------------------------------------------------------------------------
__global__ void scatter_add(const float* __restrict__ feature,
                            const int*   __restrict__ src,
                            const int*   __restrict__ dst,
                            float*       __restrict__ agg) {
  const int lane = threadIdx.x & 31;
  const int e    = blockIdx.x * 8 + (threadIdx.x >> 5);   // 8 edges / 256-thread block
  const int s = src[e];
  const int d = dst[e];
  const int f = lane * 4;
  const float4 v = *(const float4*)(feature + (size_t)s * INF + f);
  float* p = agg + (size_t)d * INF + f;
  atomicAdd(p + 0, v.x);   // global_atomic_add_f32, no-return form
  atomicAdd(p + 1, v.y);
  atomicAdd(p + 2, v.z);
  atomicAdd(p + 3, v.w);
}

// ---------------------------------------------------------------------------
// Kernel 3: out = relu(agg @ W^T + b) using V_WMMA_F32_16X16X4_F32.
// Block = 256 threads = 8 waves. Block owns one 16-row M tile; wave w owns
// N-tile columns [16w, 16w+16). A tile staged in LDS (coalesced), W read
// from global (64KB, stays hot in WGP$/L2 across all 3125 blocks).
//
// Fragment layout per ISA 7.12.2 (wave32):
//   A 16x4 f32 : lane L holds A[M=L&15][K=kb], A[M=L&15][K=kb+1], kb=2*(L>>4)
//   B 4x16 f32 : lane L holds B[K=kb][N=L&15], B[K=kb+1][N=L&15]
//   C/D 16x16  : VGPR i, lanes 0-15 -> row i, lanes 16-31 -> row i+8
// ---------------------------------------------------------------------------
__global__ void gemm_bias_relu(const float* __restrict__ agg,
                               const float* __restrict__ W,
                               const float* __restrict__ bias,
                               float*       __restrict__ out) {
  __shared__ float sA[16 * INF];   // 8 KB: 16 rows of agg

  const int m0 = blockIdx.x * 16;

  // Cooperative, fully-coalesced stage of the 16x128 A tile (512 float4s).
  {
    const float4* g = (const float4*)(agg + (size_t)m0 * INF);
    float4* l = (float4*)sA;
    l[threadIdx.x]       = g[threadIdx.x];
    l[threadIdx.x + 256] = g[threadIdx.x + 256];
  }
  __syncthreads();

  const int lane = threadIdx.x & 31;
  const int wave = threadIdx.x >> 5;
  const int n0   = wave * 16;
  const int r    = lane & 15;          // M-row for A frag / N-col for B frag
  const int kb   = (lane >> 4) * 2;    // K sub-offset per half-wave

  const float* arow = sA + r * INF + kb;                      // LDS
  const float* wrow = W + (size_t)(n0 + r) * INF + kb;        // global (B[k][n] = W[n][k])

  v8f acc = {};
#pragma unroll
  for (int k = 0; k < INF; k += 4) {
    v2f a = *(const v2f*)(arow + k);   // ds_load_b64
    v2f b = *(const v2f*)(wrow + k);   // global_load_b64
    acc = __builtin_amdgcn_wmma_f32_16x16x4_f32(
        /*neg_a=*/false, a, /*neg_b=*/false, b,
        /*c_mod=*/(short)0, acc, /*reuse_a=*/false, /*reuse_b=*/false);
  }

  // Epilogue: bias + ReLU, direct f32 store (contiguous 64B per half-wave row).
  const int   col     = n0 + r;
  const float bv      = bias[col];
  const int   rowbase = m0 + (lane >> 4) * 8;
#pragma unroll
  for (int i = 0; i < 8; ++i) {
    float v = acc[i] + bv;
    out[(size_t)(rowbase + i) * OUTF + col] = v > 0.f ? v : 0.f;
  }
}

// ---------------------------------------------------------------------------
extern "C" void kernel_launch(void* const* d_in, const int* in_sizes, int n_in,
                              void* d_out, int out_size, void* d_ws, size_t ws_size,
                              hipStream_t stream) {
  const float* feature = (const float*)d_in[0];
  const int*   src     = (const int*)  d_in[1];
  const int*   dst     = (const int*)  d_in[2];
  const float* W       = (const float*)d_in[3];
  const float* b       = (const float*)d_in[4];
  float* out = (float*)d_out;
  float* agg = (float*)d_ws;           // 50000*128*4 = 25.6 MB scratch

  // 6,400,000 floats = 1,600,000 float4 / 256 threads = 6250 blocks
  zero_ws<<<6250, 256, 0, stream>>>((float4*)agg);
  // 640,000 edges / 8 per block = 80,000 blocks
  scatter_add<<<NEDGES / 8, 256, 0, stream>>>(feature, src, dst, agg);
  // 50,000 rows / 16 per block = 3125 blocks
  gemm_bias_relu<<<NNODES / 16, 256, 0, stream>>>(agg, W, b, out);
}